// SCEmbeddingsLayer_28355374088555
// MI455X (gfx1250) — compile-verified
//
#include <hip/hip_runtime.h>

typedef float v2f __attribute__((ext_vector_type(2)));
typedef float v8f __attribute__((ext_vector_type(8)));
typedef unsigned int u32x4 __attribute__((ext_vector_type(4)));
typedef unsigned int u32x8 __attribute__((ext_vector_type(8)));

#define BB    8
#define SS    4096
#define HH    1024
#define HIDD  128
#define NSPEC 5
#define VV    60000

// dynamic LDS layout (dwords)
#define A_PITCH   130
#define A_DW      (16 * A_PITCH)        // 2080 dw : leaky activation tile
#define G_PITCH   1028                  // 1024 + 1 pad dword per 256 dwords (TDM pad)
#define G_DW      (16 * G_PITCH + 32)   // gene rows staged by TDM (+ slack)
#define SMEM_DW   (A_DW + G_DW)
#define SMEM_BYTES (SMEM_DW * 4)

__launch_bounds__(256)
__global__ void sc_embed_kernel(
    const int*   __restrict__ gene_ids,    // [B,S]
    const float* __restrict__ expr,        // [B,S]
    const float* __restrict__ gene_emb,    // [V,H]
    const float* __restrict__ W1,          // [HID]
    const float* __restrict__ b1,          // [HID]
    const float* __restrict__ W2,          // [HID,H]
    const float* __restrict__ b2,          // [H]
    const float* __restrict__ special_emb, // [NSPEC,HID]
    const float* __restrict__ pos_emb,     // [S,H]
    const float* __restrict__ gamma,       // [H]
    const float* __restrict__ beta,        // [H]
    float*       __restrict__ out)         // [B,S,H]
{
    extern __shared__ float smem[];
    float* lds_a    = smem;           // [16][130]
    float* lds_gene = smem + A_DW;    // [16][1028] (TDM-padded rows)

    __shared__ float lds_sum[16];
    __shared__ float lds_sq[16];
    __shared__ float lds_mu[16];
    __shared__ float lds_ri[16];

    const int tb   = blockIdx.x;        // 16-token tile index
    const int t    = threadIdx.x;
    const int lane = t & 31;
    const int wave = t >> 5;            // 0..7, owns N columns [wave*128, wave*128+128)
    const int tok0 = tb * 16;

    if (t < 16) { lds_sum[t] = 0.0f; lds_sq[t] = 0.0f; }

    // ------------- Wave 0: launch TDM gather of 16 gene_emb rows into LDS -------------
    if (wave == 0) {
        unsigned gg[16];
        #pragma unroll
        for (int r = 0; r < 16; ++r) gg[r] = (unsigned)gene_ids[tok0 + r];

        unsigned long long gaddr = (unsigned long long)(const void*)gene_emb;
        unsigned ldsAddr = (unsigned)(unsigned long long)(const void*)lds_gene;

        // D# group 0: count=1, gather_mode=1 (16-bit indices), lds_addr, 57b global, type=2
        u32x4 g0;
        g0[0] = 0x80000001u;                       // gather_mode<<31 | count=1
        g0[1] = ldsAddr;                           // LDS byte address
        g0[2] = (unsigned)gaddr;                   // global addr [31:0] -> bits 95:64
        g0[3] = (unsigned)((gaddr >> 32) & 0x01FFFFFFu) | (2u << 30); // addr[56:32] | type=2

        // D# group 1
        u32x8 g1;
        g1[0] = (2u << 16)          // data_size = 4 bytes
              | (1u << 20)          // pad_enable
              | (7u << 22)          // pad_interval: every 256 dwords
              | (0u << 25);         // pad_amount: 1 dword
        g1[1] = (unsigned)((HH & 0xFFFFu) << 16);          // tensor_dim0 low16 @ bits 63:48
        g1[2] = (unsigned)((HH >> 16) & 0xFFFFu)           // tensor_dim0 hi @ bits 79:64
              | (unsigned)((VV & 0xFFFFu) << 16);          // tensor_dim1 low16 @ 95:80
        g1[3] = (unsigned)((VV >> 16) & 0xFFFFu)           // tensor_dim1 hi @ 111:96
              | (unsigned)((HH & 0xFFFFu) << 16);          // tile_dim0 = 1024 @ 127:112
        g1[4] = 16u;                                       // tile_dim1 = #gather indices
        g1[5] = (unsigned)HH;                              // tensor_dim0_stride low32
        g1[6] = 0u;                                        // stride hi / dim1_stride (ignored)
        g1[7] = 0u;

        // D# groups 2/3: sixteen 16-bit row indices
        u32x4 g2, g3;
        #pragma unroll
        for (int r = 0; r < 4; ++r) {
            g2[r] = (gg[2 * r] & 0xFFFFu) | (gg[2 * r + 1] << 16);
            g3[r] = (gg[8 + 2 * r] & 0xFFFFu) | (gg[8 + 2 * r + 1] << 16);
        }

        asm volatile("tensor_load_to_lds %0, %1, %2, %3"
                     :
                     : "s"(g0), "s"(g1), "s"(g2), "s"(g3)
                     : "memory");
    }

    // ---------------- Phase 1: build A = leaky(h) in LDS ----------------
    #pragma unroll
    for (int i = 0; i < 8; ++i) {
        int idx = t + i * 256;          // 0..2047 over [16 tokens][128 hid]
        int m = idx >> 7;
        int j = idx & 127;
        float x = expr[tok0 + m];
        bool special = (x < 0.0f);
        int xi = (int)x;                // trunc toward zero, matches astype(int32)
        int mapped = -(xi + 1);
        mapped = mapped < 0 ? 0 : (mapped > NSPEC - 1 ? NSPEC - 1 : mapped);
        float c = fmaf(x, W1[j], b1[j]);
        c = (c >= 0.0f) ? c : 0.01f * c;            // leaky inside dense_1
        float h = special ? special_emb[mapped * HIDD + j] : c;
        h = (h >= 0.0f) ? h : 0.01f * h;            // leaky before dense_2
        lds_a[m * A_PITCH + j] = h;
    }
    __syncthreads();

    // ---------------- Phase 2: expr_e = A @ W2 via V_WMMA_F32_16X16X4_F32 ----------------
    v8f acc[8] = {};                     // 8 N-tiles of 16x16 f32 accumulators

    const int aRow  = lane & 15;                 // M index
    const int kHi   = (lane & 16) ? 2 : 0;       // lane-half selects K+2
    const int nLane = (lane & 15);               // N index within tile
    const int nSlab = wave * 128;

    for (int kt = 0; kt < 32; ++kt) {
        int k0 = kt * 4 + kHi;
        // A fragment: v0 = A[m][k0], v1 = A[m][k0+1]  (8B-aligned LDS read)
        v2f a = *(const v2f*)&lds_a[aRow * A_PITCH + k0];
        const float* w2r = W2 + (long)k0 * HH;
        #pragma unroll
        for (int nt = 0; nt < 8; ++nt) {
            int n = nSlab + nt * 16 + nLane;
            v2f bfrag;
            bfrag.x = w2r[n];            // B[k0  ][n]
            bfrag.y = w2r[HH + n];       // B[k0+1][n]
            acc[nt] = __builtin_amdgcn_wmma_f32_16x16x4_f32(
                false, a, false, bfrag, (short)0, acc[nt], false, false);
        }
    }

    // gene rows must be in LDS before the epilogue
    if (wave == 0) __builtin_amdgcn_s_wait_tensorcnt(0);
    __syncthreads();

    // ---------------- Phase 3: fused epilogue (bias + pos + LDS gene + LayerNorm) ----------------
    const int rOff = (lane & 16) ? 8 : 0;        // D layout: lanes 16-31 hold rows M+8
    const int s0   = (tok0 & (SS - 1)) + rOff;   // tiles never cross a batch boundary

    float rsum[8] = {0,0,0,0,0,0,0,0};
    float rsq [8] = {0,0,0,0,0,0,0,0};

    #pragma unroll
    for (int nt = 0; nt < 8; ++nt) {
        int n = nSlab + nt * 16 + nLane;
        int nAdj = n + (n >> 8);                 // TDM pad: +1 dword per 256 dwords
        float bias = b2[n];
        #pragma unroll
        for (int r = 0; r < 8; ++r) {
            float e = acc[nt][r]
                    + lds_gene[(rOff + r) * G_PITCH + nAdj]
                    + pos_emb[(long)(s0 + r) * HH + n]
                    + bias;
            acc[nt][r] = e;
            rsum[r] += e;
            rsq[r]  += e * e;
        }
    }

    // reduce across the 16 lanes of each half (xor masks < 16 stay within a half)
    #pragma unroll
    for (int off = 1; off <= 8; off <<= 1) {
        #pragma unroll
        for (int r = 0; r < 8; ++r) {
            rsum[r] += __shfl_xor(rsum[r], off, 32);
            rsq[r]  += __shfl_xor(rsq[r],  off, 32);
        }
    }
    if (nLane == 0) {                    // one lane per half contributes its 8 rows
        #pragma unroll
        for (int r = 0; r < 8; ++r) {
            atomicAdd(&lds_sum[rOff + r], rsum[r]);
            atomicAdd(&lds_sq [rOff + r], rsq[r]);
        }
    }
    __syncthreads();

    if (t < 16) {
        float mu  = lds_sum[t] * (1.0f / HH);
        float var = lds_sq[t] * (1.0f / HH) - mu * mu;
        lds_mu[t] = mu;
        lds_ri[t] = rsqrtf(var + 1e-12f);
    }
    __syncthreads();

    float mu[8], ri[8];
    #pragma unroll
    for (int r = 0; r < 8; ++r) { mu[r] = lds_mu[rOff + r]; ri[r] = lds_ri[rOff + r]; }

    #pragma unroll
    for (int nt = 0; nt < 8; ++nt) {
        int n = nSlab + nt * 16 + nLane;
        float ga = gamma[n];
        float be = beta[n];
        #pragma unroll
        for (int r = 0; r < 8; ++r) {
            long g = (long)(tok0 + rOff + r);
            out[g * HH + n] = (acc[nt][r] - mu[r]) * ri[r] * ga + be;
        }
    }
}

extern "C" void kernel_launch(void* const* d_in, const int* in_sizes, int n_in,
                              void* d_out, int out_size, void* d_ws, size_t ws_size,
                              hipStream_t stream) {
    const int*   gene_ids    = (const int*)  d_in[0];
    const float* expr        = (const float*)d_in[1];
    const float* gene_emb    = (const float*)d_in[2];
    const float* W1          = (const float*)d_in[3];
    const float* b1          = (const float*)d_in[4];
    const float* W2          = (const float*)d_in[5];
    const float* b2          = (const float*)d_in[6];
    const float* special_emb = (const float*)d_in[7];
    const float* pos_emb     = (const float*)d_in[8];
    const float* gamma       = (const float*)d_in[9];
    const float* beta        = (const float*)d_in[10];
    float* out = (float*)d_out;

    int ntok = in_sizes[0];              // B*S
    int nblocks = ntok / 16;             // 2048
    sc_embed_kernel<<<nblocks, 256, SMEM_BYTES, stream>>>(
        gene_ids, expr, gene_emb, W1, b1, W2, b2,
        special_emb, pos_emb, gamma, beta, out);
}